// S4D_48490180772523
// MI455X (gfx1250) — compile-verified
//
#include <hip/hip_runtime.h>

typedef __attribute__((ext_vector_type(2))) float v2f;
typedef __attribute__((ext_vector_type(8))) float v8f;

#define Bsz 32
#define Hsz 256
#define Lsz 4096
#define Nst 64
#define Q   64
#define NCH (Lsz / Q)   // 64 chunks
#define BG  16          // batch tile (WMMA N dim)
#define NBG (Bsz / BG)  // 2
#define MPAD 66         // row stride for 64x64 setup matrices (read once)
#define UP2  68         // row stride (floats) for [batch][row] tiles: 16B-aligned 8-float runs
#define NT   (Q / 4)    // 16 WMMA K-steps per 64-deep matmul

// y[b,h,l] = sum_{j<=l} K[h,l-j] u[b,h,j] + d_h u[b,h,l]
// K[h,m]   = Re sum_n coef_n w_n^m,  w = exp(dt*A),  coef = C*(exp(dtA)-1)/A
// Chunked scan, per chunk c of size Q:
//   Y = T@U + Vre@Sre + (-Vim)@Sim + d*U      (T Toeplitz of K, V[i,n]=coef_n w^i)
//   p = W@U (complex, W[n,i']=w^{Q-i'}),  S <- w^Q * S + p
__global__ __launch_bounds__(128)
void s4d_chunked_wmma_f32(const float* __restrict__ u,
                          const float* __restrict__ lam_re,
                          const float* __restrict__ lam_im,
                          const float* __restrict__ cc,
                          const float* __restrict__ dvec,
                          const float* __restrict__ delta,
                          float* __restrict__ y)
{
  __shared__ float Tm  [Q * MPAD];
  __shared__ float Vre [Q * MPAD];
  __shared__ float VimN[Q * MPAD];     // negated Im(V): pure accumulation in WMMA
  __shared__ float Wre [Q * MPAD];
  __shared__ float Wim [Q * MPAD];
  __shared__ float Ub  [2][BG * UP2];  // double-buffered u chunk, [batch][time]
  __shared__ float Sre [BG * UP2];     // state, [batch][state n]
  __shared__ float Sim [BG * UP2];
  __shared__ float Yb  [BG * UP2];     // [batch][time]
  __shared__ float Kv  [Q];
  __shared__ float wqr [Nst];
  __shared__ float wqi [Nst];

  const int tid  = threadIdx.x;
  const int h    = blockIdx.x >> 1;   // head
  const int bg   = blockIdx.x & 1;    // batch group
  const int lane = tid & 31;

  const float dt = __expf(delta[h]);
  const float dh = dvec[h];

  // ---------------- per-head matrix setup (LDS, one-time) ----------------
  for (int idx = tid; idx < Q * Nst; idx += 128) {
    const int i = idx >> 6;     // row
    const int n = idx & 63;     // state
    float ar = fminf(lam_re[h * Nst + n], -1e-4f);
    float ai = lam_im[h * Nst + n];
    float dre = dt * ar, dim = dt * ai;
    float er = __expf(dre);
    float cs = __cosf(dim), sn = __sinf(dim);
    float gr = er * cs - 1.0f, gi = er * sn;
    float inv = 1.0f / (ar * ar + ai * ai);
    float qr = (gr * ar + gi * ai) * inv;
    float qi = (gi * ar - gr * ai) * inv;
    float cr = cc[(h * Nst + n) * 2 + 0];
    float ci = cc[(h * Nst + n) * 2 + 1];
    float coefr = cr * qr - ci * qi;
    float coefi = cr * qi + ci * qr;
    float fi = (float)i;                        // E = w^i
    float em = __expf(dre * fi);
    float ec = __cosf(dim * fi), es = __sinf(dim * fi);
    float Er = em * ec, Ei = em * es;
    Vre [i * MPAD + n] =  coefr * Er - coefi * Ei;
    VimN[i * MPAD + n] = -(coefr * Ei + coefi * Er);
    float fm = (float)(Q - i);                  // W[n][i] = w^(Q-i)
    float wm = __expf(dre * fm);
    float wc = __cosf(dim * fm), ws = __sinf(dim * fm);
    Wre[n * MPAD + i] = wm * wc;
    Wim[n * MPAD + i] = wm * ws;
    if (i == 0) { wqr[n] = wm * wc; wqi[n] = wm * ws; }   // w^Q
  }
  __syncthreads();
  if (tid < Q) {                    // K[m] = row-sum of Vre
    float s = 0.0f;
    for (int n = 0; n < Nst; ++n) s += Vre[tid * MPAD + n];
    Kv[tid] = s;
  }
  for (int idx = tid; idx < BG * UP2; idx += 128) { Sre[idx] = 0.0f; Sim[idx] = 0.0f; }
  __syncthreads();
  for (int idx = tid; idx < Q * Q; idx += 128) {  // lower-triangular Toeplitz of K
    int i = idx >> 6, j = idx & 63;
    Tm[i * MPAD + j] = (i >= j) ? Kv[i - j] : 0.0f;
  }
  __syncthreads();

  // ---------------- hoist chunk-invariant A fragments + w^Q into registers ----------------
  // ISA 32-bit A 16x4 layout: lane -> M = lane&15; VGPR0 = K{0,2}, VGPR1 = K{1,3} by lane half
  const int mrow = (tid >> 5) * 16;         // this wave's 16-row slice
  const int rA   = lane & 15;
  const int kkA  = (lane >> 4) * 2;
  const int rowD = mrow + (lane >> 4) * 8;  // D-frag: VGPR j -> row rowD + j
  v2f aT[NT], aVr[NT], aVi[NT], aWr[NT], aWi[NT];
  #pragma unroll
  for (int t = 0; t < NT; ++t) {
    const int aoff = (mrow + rA) * MPAD + 4 * t + kkA;
    aT [t] = v2f{ Tm  [aoff], Tm  [aoff + 1] };
    aVr[t] = v2f{ Vre [aoff], Vre [aoff + 1] };
    aVi[t] = v2f{ VimN[aoff], VimN[aoff + 1] };
    aWr[t] = v2f{ Wre [aoff], Wre [aoff + 1] };
    aWi[t] = v2f{ Wim [aoff], Wim [aoff + 1] };
  }
  const float4 wr0 = *(const float4*)&wqr[rowD];       // w^Q for this lane's 8 rows
  const float4 wr1 = *(const float4*)&wqr[rowD + 4];
  const float4 wi0 = *(const float4*)&wqi[rowD];
  const float4 wi1 = *(const float4*)&wqi[rowD + 4];

  // ---------------- prologue: stage chunk 0 ----------------
  const size_t bstride = (size_t)Hsz * Lsz;
  const size_t base0   = ((size_t)(bg * BG) * Hsz + (size_t)h) * Lsz;
  const int i4 = (tid & 15) * 4;      // 4-float run along l
  const int bA = tid >> 4;            // batches 0..7 (and bA+8)

  float4 u0 = *(const float4*)(u + base0 + (size_t)bA * bstride + i4);
  float4 u1 = *(const float4*)(u + base0 + (size_t)(bA + 8) * bstride + i4);
  *(float4*)&Ub[0][bA * UP2 + i4]       = u0;
  *(float4*)&Ub[0][(bA + 8) * UP2 + i4] = u1;
  __syncthreads();

  // ---------------- chunk scan ----------------
  const int bcol = lane & 15;
  const int kkB  = (lane >> 4) * 2;
  const int eoff = bcol * UP2 + rowD;   // epilogue: this lane's 8 contiguous floats

  #pragma unroll 1
  for (int c = 0; c < NCH; ++c) {
    const float* UbC = Ub[c & 1];
    float*       UbN = Ub[(c & 1) ^ 1];
    // prefetch next chunk into registers (overlaps with compute below)
    if (c + 1 < NCH) {
      const float* up = u + base0 + (size_t)(c + 1) * Q;
      u0 = *(const float4*)(up + (size_t)bA * bstride + i4);
      u1 = *(const float4*)(up + (size_t)(bA + 8) * bstride + i4);
    }

    // 4 independent WMMA chains: yT (T@U), yS (Vre@Sre + (-Vim)@Sim), pre, pim
    v8f yT = {}, yS = {}, pre = {}, pim = {};
    #pragma unroll
    for (int t = 0; t < NT; ++t) {
      const int boff = bcol * UP2 + 4 * t + kkB;   // contiguous pair -> single b64
      v2f bU  = *(const v2f*)&UbC[boff];
      v2f bSr = *(const v2f*)&Sre[boff];
      v2f bSi = *(const v2f*)&Sim[boff];
      yT  = __builtin_amdgcn_wmma_f32_16x16x4_f32(false, aT [t], false, bU,  (short)0, yT,  false, false);
      yS  = __builtin_amdgcn_wmma_f32_16x16x4_f32(false, aVr[t], false, bSr, (short)0, yS,  false, false);
      yS  = __builtin_amdgcn_wmma_f32_16x16x4_f32(false, aVi[t], false, bSi, (short)0, yS,  false, false);
      pre = __builtin_amdgcn_wmma_f32_16x16x4_f32(false, aWr[t], false, bU,  (short)0, pre, false, false);
      pim = __builtin_amdgcn_wmma_f32_16x16x4_f32(false, aWi[t], false, bU,  (short)0, pim, false, false);
    }
    __syncthreads();   // all cross-wave reads of Sre/Sim/UbC complete

    // epilogue: y = yT + yS + d*u ; S <- w^Q*S + p   (contiguous float4 per lane)
    {
      const float4 ua  = *(const float4*)&UbC[eoff];
      const float4 ub  = *(const float4*)&UbC[eoff + 4];
      const float4 sr0 = *(const float4*)&Sre[eoff];
      const float4 sr1 = *(const float4*)&Sre[eoff + 4];
      const float4 si0 = *(const float4*)&Sim[eoff];
      const float4 si1 = *(const float4*)&Sim[eoff + 4];
      float4 yo0, yo1, nr0, nr1, ni0, ni1;
#define EPI(Cp, jj, U4, SR4, SI4, YO, NR, NI, WR4, WI4)              \
      YO.Cp = yT[jj] + yS[jj] + dh * U4.Cp;                          \
      NR.Cp = WR4.Cp * SR4.Cp - WI4.Cp * SI4.Cp + pre[jj];           \
      NI.Cp = WR4.Cp * SI4.Cp + WI4.Cp * SR4.Cp + pim[jj];
      EPI(x, 0, ua, sr0, si0, yo0, nr0, ni0, wr0, wi0)
      EPI(y, 1, ua, sr0, si0, yo0, nr0, ni0, wr0, wi0)
      EPI(z, 2, ua, sr0, si0, yo0, nr0, ni0, wr0, wi0)
      EPI(w, 3, ua, sr0, si0, yo0, nr0, ni0, wr0, wi0)
      EPI(x, 4, ub, sr1, si1, yo1, nr1, ni1, wr1, wi1)
      EPI(y, 5, ub, sr1, si1, yo1, nr1, ni1, wr1, wi1)
      EPI(z, 6, ub, sr1, si1, yo1, nr1, ni1, wr1, wi1)
      EPI(w, 7, ub, sr1, si1, yo1, nr1, ni1, wr1, wi1)
#undef EPI
      *(float4*)&Yb[eoff]      = yo0;
      *(float4*)&Yb[eoff + 4]  = yo1;
      *(float4*)&Sre[eoff]     = nr0;
      *(float4*)&Sre[eoff + 4] = nr1;
      *(float4*)&Sim[eoff]     = ni0;
      *(float4*)&Sim[eoff + 4] = ni1;
    }
    // stage prefetched chunk into the other buffer (its last readers finished at chunk c-1)
    if (c + 1 < NCH) {
      *(float4*)&UbN[bA * UP2 + i4]       = u0;
      *(float4*)&UbN[(bA + 8) * UP2 + i4] = u1;
    }
    __syncthreads();   // Yb, S, and U(c+1) visible

    // store y chunk (coalesced: 16 lanes cover 64 consecutive floats per batch row)
    {
      float* yp = y + base0 + (size_t)c * Q;
      const float4 va = *(const float4*)&Yb[bA * UP2 + i4];
      const float4 vb = *(const float4*)&Yb[(bA + 8) * UP2 + i4];
      *(float4*)(yp + (size_t)bA * bstride + i4)       = va;
      *(float4*)(yp + (size_t)(bA + 8) * bstride + i4) = vb;
    }
    // next iteration's first barrier separates these Yb reads from the next epilogue writes
  }
}

extern "C" void kernel_launch(void* const* d_in, const int* in_sizes, int n_in,
                              void* d_out, int out_size, void* d_ws, size_t ws_size,
                              hipStream_t stream) {
  const float* u      = (const float*)d_in[0];  // [B,H,L]
  const float* lam_re = (const float*)d_in[1];  // [H,N]
  const float* lam_im = (const float*)d_in[2];  // [H,N]
  const float* c      = (const float*)d_in[3];  // [H,N,2]
  const float* d      = (const float*)d_in[4];  // [H]
  const float* delta  = (const float*)d_in[5];  // [H,1]
  float* yout = (float*)d_out;                  // [B,H,L]
  (void)in_sizes; (void)n_in; (void)out_size; (void)d_ws; (void)ws_size;

  dim3 grid(Hsz * NBG);   // 512 workgroups: (head, batch-group-of-16)
  dim3 block(128);        // 4 wave32s
  s4d_chunked_wmma_f32<<<grid, block, 0, stream>>>(u, lam_re, lam_im, c, d, delta, yout);
}